// Net_37443524886758
// MI455X (gfx1250) — compile-verified
//
#include <hip/hip_runtime.h>

typedef __attribute__((ext_vector_type(2))) float v2f;
typedef __attribute__((ext_vector_type(8))) float v8f;

#define N_NODES 100000
#define R_REL   32
#define E_EDGES 3200000
#define F_INF   128
#define H_DIM   16
#define C_DIM   8

// ---------------------------------------------------------------- utilities
__global__ void zero_kernel(float* __restrict__ p, int n) {
  int i = blockIdx.x * blockDim.x + threadIdx.x;
  if (i < n) p[i] = 0.0f;
}

// deg[dst*R + etype] += 1
__global__ void deg_kernel(const int* __restrict__ ei, const int* __restrict__ et,
                           float* __restrict__ deg) {
  int e = blockIdx.x * blockDim.x + threadIdx.x;
  if (e >= E_EDGES) return;
  int dst = ei[E_EDGES + e];
  int r   = et[e];
  atomicAdd(deg + (size_t)dst * R_REL + r, 1.0f);
}

// ------------------------------------------------- layer-1 dense transform
// xw1[r, n, h] = sum_f emb[n, f] * W1[r, f, h]
// One block = 16 nodes staged in LDS, 8 waves x 4 relations each,
// K=128 via 32x V_WMMA_F32_16X16X4_F32.
__global__ __launch_bounds__(256)
void gemm1_kernel(const float* __restrict__ emb, const float* __restrict__ W1,
                  float* __restrict__ xw1) {
  __shared__ float sA[16 * 132];          // 16 rows x 128 cols, pitch 132 (bank-safe)
  const int n0 = blockIdx.x * 16;
  const int t  = threadIdx.x;

  // cooperative load: 16x128 f32 = 512 float4
  for (int i = t; i < 512; i += 256) {
    int row = i >> 5;                     // 32 float4 per row
    int c4  = i & 31;
    float4 v = ((const float4*)(emb + (size_t)(n0 + row) * F_INF))[c4];
    float* d = sA + row * 132 + c4 * 4;
    d[0] = v.x; d[1] = v.y; d[2] = v.z; d[3] = v.w;
  }
  __syncthreads();

  const int lane = t & 31;
  const int wave = t >> 5;
  const int ll = lane & 15;               // col (B/D) / row (A)
  const int lh = lane >> 4;               // half-wave
  const float* arow = sA + ll * 132 + lh * 2;

  for (int rr = 0; rr < 4; ++rr) {
    const int r = wave + rr * 8;
    const float* Br = W1 + (size_t)r * F_INF * H_DIM;   // [128][16] row-major
    v8f acc = {};
#pragma unroll 4
    for (int k0 = 0; k0 < F_INF; k0 += 4) {
      v2f a, b;
      a.x = arow[k0];                     // A: M=ll, K=k0+2*lh
      a.y = arow[k0 + 1];                 //     K=k0+2*lh+1
      const float* bp = Br + (size_t)(k0 + 2 * lh) * H_DIM + ll;
      b.x = bp[0];                        // B: K=k0+2*lh,   N=ll
      b.y = bp[H_DIM];                    //    K=k0+2*lh+1, N=ll
      acc = __builtin_amdgcn_wmma_f32_16x16x4_f32(false, a, false, b,
                                                  (short)0, acc, false, false);
    }
    // D layout: acc[v] -> (M = v + 8*lh, N = ll)
    float* out = xw1 + ((size_t)r * N_NODES + (size_t)(n0 + 8 * lh)) * H_DIM + ll;
#pragma unroll
    for (int v = 0; v < 8; ++v) out[(size_t)v * H_DIM] = acc[v];
  }
}

// --------------------------------------------------- layer-1 edge scatter
// h[dst, :] += xw1[etype, src, :] / max(deg[dst*R+etype], 1)
__global__ void scatter1_kernel(const int* __restrict__ ei, const int* __restrict__ et,
                                const float* __restrict__ deg,
                                const float* __restrict__ xw1,
                                float* __restrict__ hout) {
  int e = blockIdx.x * blockDim.x + threadIdx.x;
  if (e >= E_EDGES) return;
  int src = ei[e];
  int dst = ei[E_EDGES + e];
  int r   = et[e];
  float norm = 1.0f / fmaxf(deg[(size_t)dst * R_REL + r], 1.0f);
  const float4* row = (const float4*)(xw1 + ((size_t)r * N_NODES + src) * H_DIM);
  float* o = hout + (size_t)dst * H_DIM;
#pragma unroll
  for (int j = 0; j < 4; ++j) {
    float4 v = row[j];
    atomicAdd(o + 4 * j + 0, v.x * norm);
    atomicAdd(o + 4 * j + 1, v.y * norm);
    atomicAdd(o + 4 * j + 2, v.z * norm);
    atomicAdd(o + 4 * j + 3, v.w * norm);
  }
}

// ------------------------------------------------- layer-2 dense transform
// xw2[r, n, c] = sum_k relu(h[n, k]) * W2[r, k, c]   (K=16, C=8 padded to 16)
__global__ __launch_bounds__(256)
void gemm2_kernel(const float* __restrict__ h, const float* __restrict__ W2,
                  float* __restrict__ xw2) {
  __shared__ float sA[16 * 20];           // 16x16, pitch 20 (bank-safe)
  const int n0 = blockIdx.x * 16;
  const int t  = threadIdx.x;
  {
    int row = t >> 4, col = t & 15;       // 256 threads load 16x16 once
    float v = h[(size_t)(n0 + row) * H_DIM + col];
    sA[row * 20 + col] = v > 0.0f ? v : 0.0f;   // fused ReLU
  }
  __syncthreads();

  const int lane = t & 31;
  const int wave = t >> 5;
  const int ll = lane & 15;
  const int lh = lane >> 4;
  const float* arow = sA + ll * 20 + lh * 2;
  const bool colok = (ll < C_DIM);

  for (int rr = 0; rr < 4; ++rr) {
    const int r = wave + rr * 8;
    const float* Br = W2 + (size_t)r * H_DIM * C_DIM;   // [16][8]
    v8f acc = {};
#pragma unroll
    for (int k0 = 0; k0 < H_DIM; k0 += 4) {
      v2f a, b;
      a.x = arow[k0];
      a.y = arow[k0 + 1];
      int kr = k0 + 2 * lh;
      b.x = colok ? Br[(size_t)kr * C_DIM + ll] : 0.0f;       // pad cols 8..15
      b.y = colok ? Br[(size_t)(kr + 1) * C_DIM + ll] : 0.0f;
      acc = __builtin_amdgcn_wmma_f32_16x16x4_f32(false, a, false, b,
                                                  (short)0, acc, false, false);
    }
    if (colok) {
      float* out = xw2 + ((size_t)r * N_NODES + (size_t)(n0 + 8 * lh)) * C_DIM + ll;
#pragma unroll
      for (int v = 0; v < 8; ++v) out[(size_t)v * C_DIM] = acc[v];
    }
  }
}

// --------------------------------------------------- layer-2 edge scatter
__global__ void scatter2_kernel(const int* __restrict__ ei, const int* __restrict__ et,
                                const float* __restrict__ deg,
                                const float* __restrict__ xw2,
                                float* __restrict__ out) {
  int e = blockIdx.x * blockDim.x + threadIdx.x;
  if (e >= E_EDGES) return;
  int src = ei[e];
  int dst = ei[E_EDGES + e];
  int r   = et[e];
  float norm = 1.0f / fmaxf(deg[(size_t)dst * R_REL + r], 1.0f);
  const float4* row = (const float4*)(xw2 + ((size_t)r * N_NODES + src) * C_DIM);
  float* o = out + (size_t)dst * C_DIM;
#pragma unroll
  for (int j = 0; j < 2; ++j) {
    float4 v = row[j];
    atomicAdd(o + 4 * j + 0, v.x * norm);
    atomicAdd(o + 4 * j + 1, v.y * norm);
    atomicAdd(o + 4 * j + 2, v.z * norm);
    atomicAdd(o + 4 * j + 3, v.w * norm);
  }
}

// ------------------------------------------------------------ log_softmax
__global__ void logsoftmax_kernel(float* __restrict__ out) {
  int n = blockIdx.x * blockDim.x + threadIdx.x;
  if (n >= N_NODES) return;
  float* p = out + (size_t)n * C_DIM;
  float x[C_DIM];
  float m = -3.402823e38f;
#pragma unroll
  for (int j = 0; j < C_DIM; ++j) { x[j] = p[j]; m = fmaxf(m, x[j]); }
  float s = 0.0f;
#pragma unroll
  for (int j = 0; j < C_DIM; ++j) s += expf(x[j] - m);
  float l = logf(s);
#pragma unroll
  for (int j = 0; j < C_DIM; ++j) p[j] = x[j] - m - l;
}

// ------------------------------------------------------------------ launch
extern "C" void kernel_launch(void* const* d_in, const int* in_sizes, int n_in,
                              void* d_out, int out_size, void* d_ws, size_t ws_size,
                              hipStream_t stream) {
  const float* emb = (const float*)d_in[0];
  const float* W1  = (const float*)d_in[1];
  const float* W2  = (const float*)d_in[2];
  const int*   ei  = (const int*)d_in[3];
  const int*   et  = (const int*)d_in[4];
  float* out = (float*)d_out;

  float* deg = (float*)d_ws;                                   // N*R
  float* xw1 = deg + (size_t)N_NODES * R_REL;                  // R*N*16
  float* h   = xw1 + (size_t)R_REL * N_NODES * H_DIM;          // N*16
  float* xw2 = h + (size_t)N_NODES * H_DIM;                    // R*N*8

  const int TB = 256;
  const int eblk = (E_EDGES + TB - 1) / TB;

  zero_kernel<<<((N_NODES * R_REL) + TB - 1) / TB, TB, 0, stream>>>(deg, N_NODES * R_REL);
  zero_kernel<<<((N_NODES * H_DIM) + TB - 1) / TB, TB, 0, stream>>>(h, N_NODES * H_DIM);
  zero_kernel<<<((N_NODES * C_DIM) + TB - 1) / TB, TB, 0, stream>>>(out, N_NODES * C_DIM);

  deg_kernel<<<eblk, TB, 0, stream>>>(ei, et, deg);
  gemm1_kernel<<<N_NODES / 16, TB, 0, stream>>>(emb, W1, xw1);
  scatter1_kernel<<<eblk, TB, 0, stream>>>(ei, et, deg, xw1, h);
  gemm2_kernel<<<N_NODES / 16, TB, 0, stream>>>(h, W2, xw2);
  scatter2_kernel<<<eblk, TB, 0, stream>>>(ei, et, deg, xw2, out);
  logsoftmax_kernel<<<(N_NODES + TB - 1) / TB, TB, 0, stream>>>(out);
}